// VnnLayer_64914135711790
// MI455X (gfx1250) — compile-verified
//
#include <hip/hip_runtime.h>
#include <math.h>

// MI455X / gfx1250, wave32. Memory-bound edge-GEMM (819MB weight stream) +
// L2-resident atomic scatter + wave-per-node normalization.

typedef float v2f __attribute__((ext_vector_type(2)));
typedef float v8f __attribute__((ext_vector_type(8)));

#define DVEC 32   // feature dim D (reference: 32)

// ---------------------------------------------------------------------------
// Kernel 1: one wave (32 lanes) per edge.
//   h = x[:, src[e], :]                (B=4 rows x 32)
//   out = tanh(h @ W_e + bias_e)       via V_WMMA_F32_16X16X4_F32
//   sums[:, dst[e], :] += out  (atomics, L2-resident)
//   cnt[dst[e]] += 1
// A-frag (16x4 f32): lane m=lane&15 is matrix row M, lane-half kh=lane>>4
//   selects K-pair {0,1} vs {2,3}; VGPR0/1 = the two K values.
// B-frag (4x16 f32): lane&15 is column N, lane-half selects K-pair.
// C/D (16x16 f32, 8 VGPRs): reg r, lanes0-15 -> M=r  => batches live in
//   regs 0..3 of lanes 0..15.
// All fragment loads are hoisted ahead of the WMMA chain and issued
// unconditionally (addresses are clamped in-bounds); masking of the unused
// A rows is a branch-free v_cndmask, so no EXEC save/restore in the loop.
// ---------------------------------------------------------------------------
__global__ __launch_bounds__(256) void vnn_edge_kernel(
    const float* __restrict__ x,       // (B, N, D)
    const int*   __restrict__ ei,      // (2, E)
    const float* __restrict__ weight,  // (E, D, D)
    const float* __restrict__ bias,    // (E, D)
    float* __restrict__ sums,          // (B, N, D) accumulators
    float* __restrict__ cnt,           // (N)
    int B, int N, int E)
{
    const int lane = threadIdx.x & 31;
    const int wave = threadIdx.x >> 5;
    const int e    = blockIdx.x * 8 + wave;
    if (e >= E) return;                      // wave-uniform: EXEC stays all-1

    const int s  = ei[e];                    // src node
    const int dv = ei[E + e];                // dst node

    const int m  = lane & 15;                // A-matrix row (batch index)
    const int kh = lane >> 4;                // K-half select
    const int nl = lane & 15;                // B-matrix column
    const bool rowValid = (m < B);

    const float* __restrict__ Wp   = weight + (size_t)e * (DVEC * DVEC);
    const int mc = rowValid ? m : (B - 1);   // clamp: address always in-bounds
    const float* __restrict__ xrow = x + ((size_t)mc * N + s) * DVEC + kh * 2;

    // run-ahead prefetch on the dominant HBM stream (global_prefetch_b8)
    if (e + 16 < E)
        __builtin_prefetch(weight + (size_t)(e + 16) * (DVEC * DVEC) + lane * 8, 0, 1);

    // ---- hoisted A fragments: 8 aligned float2 loads, branch-free mask ----
    v2f af[8];
#pragma unroll
    for (int kt = 0; kt < 8; ++kt) {
        v2f t = *(const v2f*)(xrow + kt * 4);          // global_load_b64
        af[kt].x = rowValid ? t.x : 0.0f;              // v_cndmask, no branch
        af[kt].y = rowValid ? t.y : 0.0f;
    }

    // ---- hoisted B fragments: whole 4KB weight tile streamed up front ----
    v2f bf0[8], bf1[8];
#pragma unroll
    for (int kt = 0; kt < 8; ++kt) {
        const int kb = kt * 4 + kh * 2;
        bf0[kt].x = Wp[kb * DVEC + nl];
        bf0[kt].y = Wp[(kb + 1) * DVEC + nl];
        bf1[kt].x = Wp[kb * DVEC + 16 + nl];
        bf1[kt].y = Wp[(kb + 1) * DVEC + 16 + nl];
    }

    v8f acc0 = {0.f,0.f,0.f,0.f,0.f,0.f,0.f,0.f};
    v8f acc1 = {0.f,0.f,0.f,0.f,0.f,0.f,0.f,0.f};
#pragma unroll
    for (int kt = 0; kt < 8; ++kt) {                    // K = 32 in steps of 4
        acc0 = __builtin_amdgcn_wmma_f32_16x16x4_f32(false, af[kt], false, bf0[kt],
                                                     (short)0, acc0, false, false);
        acc1 = __builtin_amdgcn_wmma_f32_16x16x4_f32(false, af[kt], false, bf1[kt],
                                                     (short)0, acc1, false, false);
    }

    // Rows 0..3 (batches) are in regs 0..3 of lanes 0..15 for each tile.
    // Move tile-1 to lanes 16..31 so the whole wave scatters 32 contiguous
    // floats per batch (coalesced 128B atomic bursts), o == lane.
    const float bv = bias[(size_t)e * DVEC + lane];
#pragma unroll
    for (int r = 0; r < 4; ++r) {
        if (r < B) {
            float t1 = __shfl(acc1[r], nl, 32);        // tile1 row r, col nl
            float v  = (lane < 16) ? acc0[r] : t1;     // o = lane for both halves
            float val = tanhf(v + bv);
            atomicAdd(sums + ((size_t)r * N + dv) * DVEC + lane, val);
        }
    }
    if (lane == 0) atomicAdd(cnt + dv, 1.0f);
}

// ---------------------------------------------------------------------------
// Kernel 2: one wave per node. B=4, D=32 -> 128 values = 32 lanes x float4.
//   lane l: batch b = l>>3, features d = (l&7)*4 .. +3
//   agg = sums/max(cnt,1); state = agg . state_w + state_b (group-of-8 reduce)
//   mean/var over all 128 (full-wave reduce); bn = (agg-mean)*rsqrt(var+eps)*g+b
// ---------------------------------------------------------------------------
__global__ __launch_bounds__(256) void vnn_node_kernel(
    const float* __restrict__ sums, const float* __restrict__ cnt,
    const float* __restrict__ sw,   const float* __restrict__ sb,
    const float* __restrict__ gamma, const float* __restrict__ beta,
    float* __restrict__ outState,   // (B, N)
    float* __restrict__ outBn,      // (B, N, D)
    int B, int N)
{
    const int lane = threadIdx.x & 31;
    const int wave = threadIdx.x >> 5;
    const int n = blockIdx.x * 8 + wave;
    if (n >= N) return;

    const int b  = lane >> 3;
    const int db = (lane & 7) * 4;

    const float4 v = *(const float4*)(sums + ((size_t)b * N + n) * DVEC + db);
    const float  c   = cnt[n];
    const float  inv = 1.0f / fmaxf(c, 1.0f);
    const float a0 = v.x * inv, a1 = v.y * inv, a2 = v.z * inv, a3 = v.w * inv;

    // state = agg @ state_w + state_b  (reduce within each group of 8 lanes)
    const float4 w4 = *(const float4*)(sw + db);
    float sp = a0 * w4.x + a1 * w4.y + a2 * w4.z + a3 * w4.w;
    sp += __shfl_xor(sp, 1, 32);
    sp += __shfl_xor(sp, 2, 32);
    sp += __shfl_xor(sp, 4, 32);
    if ((lane & 7) == 0) outState[(size_t)b * N + n] = sp + sb[0];

    // mean / var over (B, D) = 128 values
    float s1 = a0 + a1 + a2 + a3;
    float s2 = a0*a0 + a1*a1 + a2*a2 + a3*a3;
#pragma unroll
    for (int off = 1; off < 32; off <<= 1) {
        s1 += __shfl_xor(s1, off, 32);
        s2 += __shfl_xor(s2, off, 32);
    }
    const float denom = (float)(B * DVEC);
    const float mean  = s1 / denom;
    const float var   = s2 / denom - mean * mean;
    const float rs    = rsqrtf(var + 1e-5f);
    const float g = gamma[n], be = beta[n];

    float4 o;
    o.x = (a0 - mean) * rs * g + be;
    o.y = (a1 - mean) * rs * g + be;
    o.z = (a2 - mean) * rs * g + be;
    o.w = (a3 - mean) * rs * g + be;
    *(float4*)(outBn + ((size_t)b * N + n) * DVEC + db) = o;
}

extern "C" void kernel_launch(void* const* d_in, const int* in_sizes, int n_in,
                              void* d_out, int out_size, void* d_ws, size_t ws_size,
                              hipStream_t stream) {
    const float* x      = (const float*)d_in[0];
    const int*   ei     = (const int*)  d_in[1];
    const float* weight = (const float*)d_in[2];
    const float* bias   = (const float*)d_in[3];
    const float* sw     = (const float*)d_in[4];
    const float* sb     = (const float*)d_in[5];
    const float* gamma  = (const float*)d_in[6];
    const float* beta   = (const float*)d_in[7];

    const int E = in_sizes[1] / 2;           // edge_index is (2, E)
    const int N = in_sizes[6];               // bn_gamma is (N,)
    const int D = in_sizes[4];               // state_w is (D, 1) -> 32
    const int B = in_sizes[0] / (N * D);     // x is (B, N, D)

    float* sums = (float*)d_ws;              // (B, N, D)
    float* cnt  = sums + (size_t)B * N * D;  // (N)
    (void)hipMemsetAsync(d_ws, 0, ((size_t)B * N * D + N) * sizeof(float), stream);

    float* outState = (float*)d_out;                 // (B, N)
    float* outBn    = outState + (size_t)B * N;      // (B, N, D)

    vnn_edge_kernel<<<(E + 7) / 8, 256, 0, stream>>>(x, ei, weight, bias,
                                                     sums, cnt, B, N, E);
    vnn_node_kernel<<<(N + 7) / 8, 256, 0, stream>>>(sums, cnt, sw, sb,
                                                     gamma, beta,
                                                     outState, outBn, B, N);
}